// NNDecoder_47751446397034
// MI455X (gfx1250) — compile-verified
//
#include <hip/hip_runtime.h>
#include <hip/hip_bf16.h>
#include <math.h>

typedef __attribute__((ext_vector_type(2))) float v2f;
typedef __attribute__((ext_vector_type(8))) float v8f;

#define IN_DIM 256
#define L1_DIM 16

// ---------------------------------------------------------------------------
// Kernel 1: node projection  h1 = relu(z @ w1_l1), h2 = relu(z @ w2_l1)
// One wave32 per 16-node tile; two f32 WMMA accumulators (one per weight mat)
// sharing the same A fragment. K-loop: 64 steps of V_WMMA_F32_16X16X4_F32.
// ---------------------------------------------------------------------------
__global__ void __launch_bounds__(128)
node_proj_wmma(const float* __restrict__ z,
               const float* __restrict__ w1,   // [256,16] row-major
               const float* __restrict__ w2,   // [256,16] row-major
               float* __restrict__ h1,         // [n_nodes,16]
               float* __restrict__ h2,         // [n_nodes,16]
               int n_nodes) {
    // Stage both weight matrices in LDS: 2 * 256*16 f32 = 32 KB.
    __shared__ float sw1[IN_DIM * L1_DIM];
    __shared__ float sw2[IN_DIM * L1_DIM];
    const int tid = threadIdx.x;                 // 0..127 (4 waves)
    for (int i = tid * 4; i < IN_DIM * L1_DIM; i += 128 * 4) {
        *(float4*)&sw1[i] = *(const float4*)&w1[i];
        *(float4*)&sw2[i] = *(const float4*)&w2[i];
    }
    __syncthreads();

    const int wave = tid >> 5;
    const int lane = tid & 31;
    const int tile = blockIdx.x * 4 + wave;
    const int row0 = tile * 16;
    if (row0 >= n_nodes) return;                 // wave-uniform guard (EXEC stays all-1)

    const int half = lane >> 4;                  // 0: lanes 0-15, 1: lanes 16-31
    const int l16  = lane & 15;

    // A fragment base: row = row0+l16, K offset half*2 (VGPR0=K0/K2, VGPR1=K1/K3)
    const float* zrow = z + (size_t)(row0 + l16) * IN_DIM + half * 2;

    v8f acc1 = {};
    v8f acc2 = {};

    #pragma unroll 4
    for (int k0 = 0; k0 < IN_DIM; k0 += 4) {
        v2f a = *(const v2f*)(zrow + k0);        // contiguous pair, 8B aligned

        // B fragment: VGPR0 = row (k0 + 2*half), VGPR1 = row (k0+1 + 2*half)
        const int kb = (k0 + 2 * half) * L1_DIM + l16;
        v2f b1, b2;
        b1.x = sw1[kb];           b1.y = sw1[kb + L1_DIM];
        b2.x = sw2[kb];           b2.y = sw2[kb + L1_DIM];

        // 8-arg form: (neg_a, A, neg_b, B, c_mod, C, reuse_a, reuse_b)
        acc1 = __builtin_amdgcn_wmma_f32_16x16x4_f32(false, a, false, b1,
                                                     (short)0, acc1, false, false);
        acc2 = __builtin_amdgcn_wmma_f32_16x16x4_f32(false, a, false, b2,
                                                     (short)0, acc2, false, false);
    }

    // C/D layout: VGPR r -> row (row0 + r + half*8), col l16
    #pragma unroll
    for (int r = 0; r < 8; ++r) {
        const int row = row0 + r + half * 8;
        const size_t o = (size_t)row * L1_DIM + l16;
        float v1 = acc1[r]; v1 = v1 > 0.0f ? v1 : 0.0f;
        float v2 = acc2[r]; v2 = v2 > 0.0f ? v2 : 0.0f;
        h1[o] = v1;
        h2[o] = v2;
    }
}

// ---------------------------------------------------------------------------
// Kernel 2: per-edge scoring (memory-bound gather).
// out[e] = sigmoid( dot(h1[src], w1_l2[et]) + dot(h2[dst], w2_l2[et]) )
// h1/h2 (3.2 MB each) and the 64 KB type tables are L2/WGP$-resident.
// ---------------------------------------------------------------------------
__global__ void __launch_bounds__(256)
edge_score(const int* __restrict__ edge_index,  // [2, n_edges]
           const int* __restrict__ edge_type,   // [n_edges]
           const float* __restrict__ h1,
           const float* __restrict__ h2,
           const float* __restrict__ w1_l2,     // [n_etype,16]
           const float* __restrict__ w2_l2,     // [n_etype,16]
           float* __restrict__ out,
           int n_edges) {
    const int e = blockIdx.x * blockDim.x + threadIdx.x;
    if (e >= n_edges) return;

    const int src = edge_index[e];
    const int dst = edge_index[n_edges + e];
    const int t   = edge_type[e];

    const float4* A = (const float4*)(h1    + (size_t)src * L1_DIM);
    const float4* B = (const float4*)(w1_l2 + (size_t)t   * L1_DIM);
    const float4* C = (const float4*)(h2    + (size_t)dst * L1_DIM);
    const float4* D = (const float4*)(w2_l2 + (size_t)t   * L1_DIM);

    float s = 0.0f;
    #pragma unroll
    for (int i = 0; i < 4; ++i) {
        const float4 a = A[i], b = B[i], c = C[i], d = D[i];
        s = fmaf(a.x, b.x, s); s = fmaf(a.y, b.y, s);
        s = fmaf(a.z, b.z, s); s = fmaf(a.w, b.w, s);
        s = fmaf(c.x, d.x, s); s = fmaf(c.y, d.y, s);
        s = fmaf(c.z, d.z, s); s = fmaf(c.w, d.w, s);
    }
    out[e] = 1.0f / (1.0f + __expf(-s));
}

// ---------------------------------------------------------------------------
// Launch
// ---------------------------------------------------------------------------
extern "C" void kernel_launch(void* const* d_in, const int* in_sizes, int n_in,
                              void* d_out, int out_size, void* d_ws, size_t ws_size,
                              hipStream_t stream) {
    const float* z          = (const float*)d_in[0];  // [n_nodes, 256]
    const int*   edge_index = (const int*)  d_in[1];  // [2, n_edges]
    const int*   edge_type  = (const int*)  d_in[2];  // [n_edges]
    const float* w1_l1      = (const float*)d_in[3];  // [256,16]
    const float* w1_l2      = (const float*)d_in[4];  // [n_etype,16]
    const float* w2_l1      = (const float*)d_in[5];  // [256,16]
    const float* w2_l2      = (const float*)d_in[6];  // [n_etype,16]
    float*       out        = (float*)d_out;

    const int n_nodes = in_sizes[0] / IN_DIM;
    const int n_edges = in_sizes[2];

    float* h1 = (float*)d_ws;
    float* h2 = h1 + (size_t)n_nodes * L1_DIM;

    // Phase 1: node projections via WMMA. One wave per 16-row tile, 4 waves/block.
    const int tiles  = (n_nodes + 15) / 16;
    const int blocks = (tiles + 3) / 4;
    node_proj_wmma<<<blocks, 128, 0, stream>>>(z, w1_l1, w2_l1, h1, h2, n_nodes);

    // Phase 2: edge gather + score.
    const int eblocks = (n_edges + 255) / 256;
    edge_score<<<eblocks, 256, 0, stream>>>(edge_index, edge_type, h1, h2,
                                            w1_l2, w2_l2, out, n_edges);
}